// CrossAttention_S_67808943669433
// MI455X (gfx1250) — compile-verified
//
#include <hip/hip_runtime.h>
#include <math.h>

typedef float v2f __attribute__((ext_vector_type(2)));
typedef float v8f __attribute__((ext_vector_type(8)));
typedef int   v4i __attribute__((ext_vector_type(4)));

#define DIM 48
#define HEADS 8
#define CHD 6
#define BATCH 4
#define IMGH 256
#define IMGW 256
#define NPIX (IMGH * IMGW)
#define FEPS 1e-12f

// ---- Pass A tiling: 32x8 interior pixels, 34x10 halo for dwconv3x3 ----
#define TW 32
#define TH 8
#define HW_ 34
#define HH_ 10
#define HPX (HW_ * HH_)   // 340
#define NPAD 352          // 340 padded to 22 N-tiles of 16

// ---- Pass C tiling ----
#define CPIX 512

// ---- CDNA5 async global<->LDS path (ASYNCcnt-tracked) ------------------
#if __has_builtin(__builtin_amdgcn_global_load_async_to_lds_b128) && \
    __has_builtin(__builtin_amdgcn_global_store_async_from_lds_b128)
#define HAVE_ASYNC_LDS 1
typedef __attribute__((address_space(1))) v4i gv4i;   // global int4*
typedef __attribute__((address_space(3))) v4i lv4i;   // LDS int4*
#else
#define HAVE_ASYNC_LDS 0
#endif

__device__ __forceinline__ void wait_async0()
{
#if __has_builtin(__builtin_amdgcn_s_wait_asynccnt)
    __builtin_amdgcn_s_wait_asynccnt(0);
#else
    asm volatile("s_wait_asynccnt 0" ::: "memory");
#endif
}

// -------------------------------------------------------------------------
// f32 WMMA 16x16x4 GEMM over an LDS tile:
//   Y[48][npad] = W[rowBase..rowBase+48)[48] * X[48][npad]
// A-frag (16x4): lanes 0-15 hold (K0,K1) of row M=lane; lanes 16-31 (K2,K3)
// B-frag (4x16): lanes 0-15 hold (K0,K1) of col N=lane; lanes 16-31 (K2,K3)
// D (16x16 f32): lanes 0-15 rows 0..7 col=lane; lanes 16-31 rows 8..15
// -------------------------------------------------------------------------
__device__ __forceinline__ void gemm48_wmma(const float* __restrict__ W, int rowBase,
                                            const float* __restrict__ X,
                                            float* __restrict__ Y,
                                            int npad, int ntiles,
                                            int wave, int lane)
{
    const int l15  = lane & 15;
    const int koff = (lane >> 4) * 2;
    for (int t = wave; t < 3 * ntiles; t += 8) {
        const int mt = t % 3;
        const int nt = t / 3;
        v8f acc = {0.f, 0.f, 0.f, 0.f, 0.f, 0.f, 0.f, 0.f};
        const int row = rowBase + mt * 16 + l15;
        const int col = nt * 16 + l15;
        #pragma unroll
        for (int k0 = 0; k0 < DIM; k0 += 4) {
            v2f a, b;
            a.x = W[row * DIM + k0 + koff];
            a.y = W[row * DIM + k0 + koff + 1];
            b.x = X[(k0 + koff) * npad + col];
            b.y = X[(k0 + koff + 1) * npad + col];
            acc = __builtin_amdgcn_wmma_f32_16x16x4_f32(
                false, a, false, b, (short)0, acc, false, false);
        }
        const int mbase = mt * 16 + (lane >> 4) * 8;
        #pragma unroll
        for (int i = 0; i < 8; ++i)
            Y[(mbase + i) * npad + col] = acc[i];
    }
}

// Stage one 34x10 halo tile (48 ch) of a [48][256][256] plane into LDS
// [48][NPAD], zero-filling outside the image (zero-pad == dwconv padding
// since conv1x1 has no bias) and in the NPAD tail.
__device__ __forceinline__ void stage_fea(float* __restrict__ stage,
                                          const float* __restrict__ src,
                                          int x0, int y0, int tid)
{
    for (int i = tid; i < DIM * NPAD; i += 256) {
        const int c = i / NPAD;
        const int p = i - c * NPAD;
        float v = 0.f;
        if (p < HPX) {
            const int y  = p / HW_;
            const int x  = p - y * HW_;
            const int gy = y0 - 1 + y;
            const int gx = x0 - 1 + x;
            if ((unsigned)gy < IMGH && (unsigned)gx < IMGW)
                v = src[c * NPIX + gy * IMGW + gx];
        }
        stage[i] = v;
    }
}

// Depthwise 3x3 over the halo buffer; one thread = one interior pixel,
// all 48 channels. out layout: [48][256] (tile-pixel-major within channel).
__device__ __forceinline__ void dwconv_tile(const float* __restrict__ bufc,
                                            const float* __restrict__ dww,
                                            float* __restrict__ out, int tid)
{
    const int py = tid >> 5;
    const int px = tid & 31;
    for (int c = 0; c < DIM; ++c) {
        const float* wp = dww + c * 9;
        const float* bp = bufc + c * NPAD + py * HW_ + px;
        float s = 0.f;
        #pragma unroll
        for (int dy = 0; dy < 3; ++dy)
            #pragma unroll
            for (int dx = 0; dx < 3; ++dx)
                s = fmaf(bp[dy * HW_ + dx], wp[dy * 3 + dx], s);
        out[c * 256 + tid] = s;
    }
}

// =========================================================================
// Pass A: fused conv1x1(WMMA) + dwconv3x3 for q,k,v.  q,k are consumed
// on-chip into Gram-matrix + norm partial sums (global atomics); only v
// is written to HBM (async-from-LDS B128).
// =========================================================================
__global__ __launch_bounds__(256) void fused_qkv_kernel(
    const float* __restrict__ fea0, const float* __restrict__ fea1,
    const float* __restrict__ qk_w, const float* __restrict__ qk_dw,
    const float* __restrict__ v_w,  const float* __restrict__ v_dw,
    float* __restrict__ v_ws, float* __restrict__ G_acc,
    float* __restrict__ qn_acc, float* __restrict__ kn_acc)
{
    extern __shared__ float smemA[];
    float* w_qk  = smemA;                    // 96*48 = 4608
    float* w_v   = w_qk  + 96 * DIM;         // 48*48 = 2304
    float* dwqk  = w_v   + DIM * DIM;        // 96*9  = 864
    float* dwv   = dwqk  + 96 * 9;           // 48*9  = 432
    float* stage = dwv   + DIM * 9;          // 48*352 = 16896
    float* bufc  = stage + DIM * NPAD;       // 48*352 = 16896
    float* qtile = bufc  + DIM * NPAD;       // 48*256 = 12288
    float* ktile = qtile + DIM * 256;        // 48*256 = 12288
    float* vtile = qtile;                    // reuse after stats

    const int tid  = threadIdx.x;
    const int lane = tid & 31;
    const int wave = tid >> 5;
    const int b    = blockIdx.z;
    const int x0   = blockIdx.x * TW;
    const int y0   = blockIdx.y * TH;

    // weights -> LDS, fea0 halo -> LDS
    for (int i = tid; i < 96 * DIM; i += 256)  w_qk[i] = qk_w[i];
    for (int i = tid; i < DIM * DIM; i += 256) w_v[i]  = v_w[i];
    for (int i = tid; i < 96 * 9; i += 256)    dwqk[i] = qk_dw[i];
    for (int i = tid; i < DIM * 9; i += 256)   dwv[i]  = v_dw[i];
    stage_fea(stage, fea0 + (long)b * DIM * NPIX, x0, y0, tid);
    __syncthreads();

    // q = dwconv(conv1x1 rows 0..47)
    gemm48_wmma(w_qk, 0, stage, bufc, NPAD, NPAD / 16, wave, lane);
    __syncthreads();
    dwconv_tile(bufc, dwqk, qtile, tid);
    __syncthreads();

    // k = dwconv(conv1x1 rows 48..95)
    gemm48_wmma(w_qk, DIM, stage, bufc, NPAD, NPAD / 16, wave, lane);
    __syncthreads();
    dwconv_tile(bufc, dwqk + DIM * 9, ktile, tid);
    stage_fea(stage, fea1 + (long)b * DIM * NPIX, x0, y0, tid);  // prep v input
    __syncthreads();

    // Gram matrix + norm partial sums over this tile's 256 pixels
    if (tid < HEADS * CHD * CHD) {  // 288 entries
        const int h = tid / 36, r = tid - h * 36, c = r / 6, d = r - c * 6;
        const float* qp = qtile + (h * CHD + c) * 256;
        const float* kp = ktile + (h * CHD + d) * 256;
        float s = 0.f;
        for (int p = 0; p < 256; ++p) s = fmaf(qp[p], kp[p], s);
        atomicAdd(&G_acc[b * 288 + tid], s);
    }
    if (tid < 2 * DIM) {  // 96 norm rows
        const float* sp = (tid < DIM) ? (qtile + tid * 256)
                                      : (ktile + (tid - DIM) * 256);
        float s = 0.f;
        for (int p = 0; p < 256; ++p) s = fmaf(sp[p], sp[p], s);
        if (tid < DIM) atomicAdd(&qn_acc[b * DIM + tid], s);
        else           atomicAdd(&kn_acc[b * DIM + tid - DIM], s);
    }
    __syncthreads();   // qtile free after this point (vtile aliases it)

    // v = dwconv(conv1x1(fea1))
    gemm48_wmma(w_v, 0, stage, bufc, NPAD, NPAD / 16, wave, lane);
    __syncthreads();
    dwconv_tile(bufc, dwv, vtile, tid);
    __syncthreads();

    // v tile -> HBM
    float* vdst = v_ws + (long)b * DIM * NPIX;
#if HAVE_ASYNC_LDS
    // async-from-LDS, 16B chunks: tile row = 32 floats = 8 chunks
    for (int i = tid; i < DIM * 256 / 4; i += 256) {   // 12 iterations
        const int c  = i >> 6;
        const int p4 = i & 63;
        const int gy = y0 + (p4 >> 3);
        const int gx = x0 + (p4 & 7) * 4;
        __builtin_amdgcn_global_store_async_from_lds_b128(
            (gv4i*)(vdst + (long)c * NPIX + gy * IMGW + gx),
            (lv4i*)(vtile + c * 256 + p4 * 4), 0, 0);
    }
    wait_async0();
#else
    for (int i = tid; i < DIM * 256; i += 256) {
        const int c  = i >> 8;
        const int p  = i & 255;
        const int gy = y0 + (p >> 5);
        const int gx = x0 + (p & 31);
        vdst[(long)c * NPIX + gy * IMGW + gx] = vtile[i];
    }
#endif
}

// =========================================================================
// Pass B: norms + softmax(6x6 per head) and fold proj:  M = proj_w @ A
// (A = block-diag of per-head attn). One tiny workgroup.
// =========================================================================
__global__ __launch_bounds__(256) void attn_proj_kernel(
    const float* __restrict__ G_acc, const float* __restrict__ qn_acc,
    const float* __restrict__ kn_acc, const float* __restrict__ temperature,
    const float* __restrict__ proj_w, float* __restrict__ M_ws)
{
    __shared__ float attnS[BATCH * HEADS * CHD * CHD];  // 1152
    const int tid = threadIdx.x;

    if (tid < BATCH * HEADS * CHD) {  // 192 softmax rows
        const int b = tid / DIM, r = tid % DIM, h = r / CHD, c = r % CHD;
        const float qs = 1.f / fmaxf(sqrtf(qn_acc[b * DIM + h * CHD + c]), FEPS);
        const float tp = temperature[h];
        float vals[CHD];
        float m = -1e30f;
        #pragma unroll
        for (int d = 0; d < CHD; ++d) {
            const float ks = 1.f / fmaxf(sqrtf(kn_acc[b * DIM + h * CHD + d]), FEPS);
            const float a  = G_acc[b * 288 + h * 36 + c * CHD + d] * qs * ks * tp;
            vals[d] = a;
            m = fmaxf(m, a);
        }
        float s = 0.f;
        #pragma unroll
        for (int d = 0; d < CHD; ++d) { vals[d] = __expf(vals[d] - m); s += vals[d]; }
        const float inv = 1.f / s;
        #pragma unroll
        for (int d = 0; d < CHD; ++d)
            attnS[b * 288 + h * 36 + c * CHD + d] = vals[d] * inv;
    }
    __syncthreads();

    // M[b][o][dglob] = sum_{c in head(dglob)} proj_w[o][h*6+c] * attn[h][c][d]
    for (int e = tid; e < BATCH * DIM * DIM; e += 256) {
        const int b = e / (DIM * DIM);
        const int r = e % (DIM * DIM);
        const int o = r / DIM, dg = r % DIM;
        const int h = dg / CHD, dl = dg % CHD;
        float s = 0.f;
        #pragma unroll
        for (int cl = 0; cl < CHD; ++cl)
            s = fmaf(proj_w[o * DIM + h * CHD + cl],
                     attnS[b * 288 + h * 36 + cl * CHD + dl], s);
        M_ws[e] = s;
    }
}

// =========================================================================
// Pass C: out = M[b] @ v   (48x48 x 48xN GEMM via f32 WMMA, 512 px/WG)
// v staged in via async-to-LDS B128; result stored via async-from-LDS B128.
// =========================================================================
__global__ __launch_bounds__(256) void out_proj_kernel(
    const float* __restrict__ v_ws, const float* __restrict__ M_ws,
    float* __restrict__ out)
{
    extern __shared__ float smemC[];
    float* Mw     = smemC;                  // 48*48 = 2304
    float* vstage = Mw + DIM * DIM;         // 48*512 = 24576
    float* obuf   = vstage + DIM * CPIX;    // 48*512 = 24576

    const int tid  = threadIdx.x;
    const int lane = tid & 31;
    const int wave = tid >> 5;
    const int b    = blockIdx.y;
    const long n0  = (long)blockIdx.x * CPIX;

    const float* vsrc = v_ws + (long)b * DIM * NPIX;
#if HAVE_ASYNC_LDS
    for (int i = tid; i < DIM * CPIX / 4; i += 256) {   // 24 iterations
        const int c  = i >> 7;                          // 128 chunks per channel
        const int p4 = i & 127;
        __builtin_amdgcn_global_load_async_to_lds_b128(
            (gv4i*)(vsrc + (long)c * NPIX + n0 + p4 * 4),
            (lv4i*)(vstage + c * CPIX + p4 * 4), 0, 0);
    }
#else
    for (int i = tid; i < DIM * CPIX; i += 256) {
        const int c = i >> 9;
        const int p = i & (CPIX - 1);
        vstage[i] = vsrc[(long)c * NPIX + n0 + p];
    }
#endif
    for (int i = tid; i < DIM * DIM; i += 256) Mw[i] = M_ws[b * DIM * DIM + i];
#if HAVE_ASYNC_LDS
    wait_async0();
#endif
    __syncthreads();

    gemm48_wmma(Mw, 0, vstage, obuf, CPIX, CPIX / 16, wave, lane);
    __syncthreads();

    float* dst = out + (long)b * DIM * NPIX;
#if HAVE_ASYNC_LDS
    for (int i = tid; i < DIM * CPIX / 4; i += 256) {   // 24 iterations
        const int c  = i >> 7;
        const int p4 = i & 127;
        __builtin_amdgcn_global_store_async_from_lds_b128(
            (gv4i*)(dst + (long)c * NPIX + n0 + p4 * 4),
            (lv4i*)(obuf + c * CPIX + p4 * 4), 0, 0);
    }
    wait_async0();
#else
    for (int i = tid; i < DIM * CPIX; i += 256) {
        const int c = i >> 9;
        const int p = i & (CPIX - 1);
        dst[(long)c * NPIX + n0 + p] = obuf[i];
    }
#endif
}

// =========================================================================
extern "C" void kernel_launch(void* const* d_in, const int* in_sizes, int n_in,
                              void* d_out, int out_size, void* d_ws, size_t ws_size,
                              hipStream_t stream)
{
    const float* fea0   = (const float*)d_in[0];
    const float* fea1   = (const float*)d_in[1];
    const float* qk_w   = (const float*)d_in[2];
    const float* qk_dw  = (const float*)d_in[3];
    const float* v_w    = (const float*)d_in[4];
    const float* v_dw   = (const float*)d_in[5];
    const float* proj_w = (const float*)d_in[6];
    const float* temp   = (const float*)d_in[7];
    float* out = (float*)d_out;

    // workspace layout
    char* ws = (char*)d_ws;
    float* v_ws = (float*)ws;                                   // 48 MB
    size_t voff = (size_t)BATCH * DIM * NPIX * sizeof(float);
    float* G_acc = (float*)(ws + voff);                         // 1152
    float* qn    = G_acc + BATCH * 288;                         // 192
    float* kn    = qn + BATCH * DIM;                            // 192
    float* M_ws  = kn + BATCH * DIM;                            // 9216

    // zero the atomic accumulators (graph-capture-safe)
    (void)hipMemsetAsync(G_acc, 0,
                         (size_t)(BATCH * 288 + 2 * BATCH * DIM) * sizeof(float),
                         stream);

    const size_t smemA =
        (size_t)(96 * DIM + DIM * DIM + 96 * 9 + DIM * 9 +
                 2 * DIM * NPAD + 2 * DIM * 256) * sizeof(float);   // ~260 KB
    dim3 gA(IMGW / TW, IMGH / TH, BATCH);
    fused_qkv_kernel<<<gA, 256, smemA, stream>>>(
        fea0, fea1, qk_w, qk_dw, v_w, v_dw, v_ws, G_acc, qn, kn);

    attn_proj_kernel<<<dim3(1), 256, 0, stream>>>(G_acc, qn, kn, temp, proj_w, M_ws);

    const size_t smemC = (size_t)(DIM * DIM + 2 * DIM * CPIX) * sizeof(float); // ~201 KB
    dim3 gC(NPIX / CPIX, BATCH);
    out_proj_kernel<<<gC, 256, smemC, stream>>>(v_ws, M_ws, out);
}